// GraphSageEmbeddingUnsup_75411035783775
// MI455X (gfx1250) — compile-verified
//
#include <hip/hip_runtime.h>
#include <hip/hip_bf16.h>

typedef __attribute__((ext_vector_type(2))) float v2f;
typedef __attribute__((ext_vector_type(8))) float v8f;

#define D 128
#define LN_EPS 1e-12f

// ---------------------------------------------------------------------------
// Kernel 1: h = LayerNorm(syn[x0] + lem[x3] + pos[x1] + sen[x2]) * g + b
// one wave per node, 4 floats per lane (128 = 32 lanes * 4)
// ---------------------------------------------------------------------------
__global__ void embed_ln_kernel(const int* __restrict__ x,
                                const float* __restrict__ syn,
                                const float* __restrict__ lem,
                                const float* __restrict__ pos,
                                const float* __restrict__ sen,
                                const float* __restrict__ g,
                                const float* __restrict__ b,
                                float* __restrict__ h, int n_nodes) {
    int wid  = (int)(blockIdx.x * (blockDim.x >> 5) + (threadIdx.x >> 5));
    int lane = threadIdx.x & 31;
    if (wid >= n_nodes) return;

    int i0 = x[wid * 4 + 0];
    int i1 = x[wid * 4 + 1];
    int i2 = x[wid * 4 + 2];
    int i3 = x[wid * 4 + 3];
    int d0 = lane * 4;

    const float4* p0 = (const float4*)(syn + (size_t)i0 * D + d0);
    const float4* p1 = (const float4*)(lem + (size_t)i3 * D + d0);
    const float4* p2 = (const float4*)(pos + (size_t)i1 * D + d0);
    const float4* p3 = (const float4*)(sen + (size_t)i2 * D + d0);
    float4 a = *p0, c = *p1, e = *p2, f = *p3;
    float4 v;
    v.x = a.x + c.x + e.x + f.x;
    v.y = a.y + c.y + e.y + f.y;
    v.z = a.z + c.z + e.z + f.z;
    v.w = a.w + c.w + e.w + f.w;

    // mean
    float s = v.x + v.y + v.z + v.w;
    for (int off = 16; off >= 1; off >>= 1) s += __shfl_xor(s, off, 32);
    float mu = s * (1.0f / (float)D);

    float4 dm;
    dm.x = v.x - mu; dm.y = v.y - mu; dm.z = v.z - mu; dm.w = v.w - mu;
    float q = dm.x * dm.x + dm.y * dm.y + dm.z * dm.z + dm.w * dm.w;
    for (int off = 16; off >= 1; off >>= 1) q += __shfl_xor(q, off, 32);
    float rs = rsqrtf(q * (1.0f / (float)D) + LN_EPS);

    float4 gg = *(const float4*)(g + d0);
    float4 bb = *(const float4*)(b + d0);
    float4 o;
    o.x = dm.x * rs * gg.x + bb.x;
    o.y = dm.y * rs * gg.y + bb.y;
    o.z = dm.z * rs * gg.z + bb.z;
    o.w = dm.w * rs * gg.w + bb.w;
    *(float4*)(h + (size_t)wid * D + d0) = o;
}

// ---------------------------------------------------------------------------
// Zero scratch
// ---------------------------------------------------------------------------
__global__ void zero_kernel(float* __restrict__ p, long long n) {
    long long t = (long long)blockIdx.x * blockDim.x + threadIdx.x;
    if (t < n) p[t] = 0.0f;
}

// ---------------------------------------------------------------------------
// Kernel 2: scatter-add  agg[dst] += h[src]  (one wave per edge, float4/lane)
// optional degree count (graph is identical for both layers -> compute once)
// ---------------------------------------------------------------------------
__global__ void scatter_kernel(const float* __restrict__ h,
                               const int* __restrict__ src,
                               const int* __restrict__ dst,
                               float* __restrict__ agg,
                               float* __restrict__ cnt,
                               int n_edges, int add_count) {
    int e    = (int)(blockIdx.x * (blockDim.x >> 5) + (threadIdx.x >> 5));
    int lane = threadIdx.x & 31;
    if (e >= n_edges) return;
    int s = src[e];
    int d = dst[e];
    float4 v = *(const float4*)(h + (size_t)s * D + lane * 4);
    float* ap = agg + (size_t)d * D + lane * 4;
    atomicAdd(ap + 0, v.x);
    atomicAdd(ap + 1, v.y);
    atomicAdd(ap + 2, v.z);
    atomicAdd(ap + 3, v.w);
    if (add_count && lane == 0) atomicAdd(cnt + d, 1.0f);
}

// ---------------------------------------------------------------------------
// Kernel 3: agg[n][d] *= 1/max(cnt[n],1)
// ---------------------------------------------------------------------------
__global__ void normalize_kernel(float* __restrict__ agg,
                                 const float* __restrict__ cnt,
                                 long long total) {
    long long t = (long long)blockIdx.x * blockDim.x + threadIdx.x;
    if (t >= total) return;
    int n = (int)(t >> 7);
    float c = cnt[n];
    agg[t] *= __frcp_rn(fmaxf(c, 1.0f));
}

// ---------------------------------------------------------------------------
// Kernel 4: out = relu?( agg @ Wl^T + bl + h @ Wr^T )
// fp32 WMMA 16x16x4. Block = 256 threads = 8 waves; wave w owns the 16x16
// tile at (rows blockIdx.x*16, cols w*16). K = 128 -> 32 steps of 4, two
// matmuls fused into one accumulator -> 64 v_wmma per tile.
//
// A 16x4 layout (2 VGPRs): lanes 0-15 hold (M=lane, K=k+0 / k+1),
//                          lanes 16-31 hold (M=lane-16, K=k+2 / k+3).
// B 4x16 mirrored; C 16x16: VGPR j = row j (lanes 0-15) / row j+8 (16-31).
// ---------------------------------------------------------------------------
__global__ void sage_gemm_wmma_kernel(const float* __restrict__ agg,
                                      const float* __restrict__ h,
                                      const float* __restrict__ Wl,
                                      const float* __restrict__ bl,
                                      const float* __restrict__ Wr,
                                      float* __restrict__ out,
                                      int do_relu) {
    int wid  = threadIdx.x >> 5;          // 0..7 -> column tile
    int lane = threadIdx.x & 31;
    int row0 = blockIdx.x * 16;
    int col0 = wid * 16;

    int ml   = lane & 15;                 // M (for A) / N (for B) within tile
    int koff = (lane >> 4) << 1;          // 0 for lanes 0-15, 2 for 16-31

    const float* aRow  = agg + (size_t)(row0 + ml) * D;
    const float* hRow  = h   + (size_t)(row0 + ml) * D;
    const float* wlRow = Wl  + (size_t)(col0 + ml) * D;   // B = Wl^T
    const float* wrRow = Wr  + (size_t)(col0 + ml) * D;   // B = Wr^T

    v8f c = {0.f, 0.f, 0.f, 0.f, 0.f, 0.f, 0.f, 0.f};

    #pragma unroll 4
    for (int k = 0; k < D; k += 4) {
        v2f a0, b0, a1, b1;
        a0.x = aRow[k + koff];  a0.y = aRow[k + koff + 1];
        b0.x = wlRow[k + koff]; b0.y = wlRow[k + koff + 1];
        c = __builtin_amdgcn_wmma_f32_16x16x4_f32(false, a0, false, b0,
                                                  (short)0, c, false, false);
        a1.x = hRow[k + koff];  a1.y = hRow[k + koff + 1];
        b1.x = wrRow[k + koff]; b1.y = wrRow[k + koff + 1];
        c = __builtin_amdgcn_wmma_f32_16x16x4_f32(false, a1, false, b1,
                                                  (short)0, c, false, false);
    }

    int colN  = col0 + ml;
    float bias = bl[colN];
    int rbase = row0 + ((lane >> 4) << 3);   // +8 for upper half lanes
    #pragma unroll
    for (int j = 0; j < 8; ++j) {
        float v = c[j] + bias;
        if (do_relu) v = fmaxf(v, 0.0f);
        out[(size_t)(rbase + j) * D + colN] = v;
    }
}

// ---------------------------------------------------------------------------
// Host-side launch
// ---------------------------------------------------------------------------
extern "C" void kernel_launch(void* const* d_in, const int* in_sizes, int n_in,
                              void* d_out, int out_size, void* d_ws, size_t ws_size,
                              hipStream_t stream) {
    const int*   x       = (const int*)  d_in[0];
    const int*   edge    = (const int*)  d_in[1];
    const float* syn     = (const float*)d_in[2];
    const float* lem     = (const float*)d_in[3];
    const float* pos     = (const float*)d_in[4];
    const float* sen     = (const float*)d_in[5];
    const float* ln_g    = (const float*)d_in[6];
    const float* ln_b    = (const float*)d_in[7];
    const float* Wl0     = (const float*)d_in[8];
    const float* bl0     = (const float*)d_in[9];
    const float* Wr0     = (const float*)d_in[10];
    const float* Wl1     = (const float*)d_in[11];
    const float* bl1     = (const float*)d_in[12];
    const float* Wr1     = (const float*)d_in[13];

    const int N = in_sizes[0] / 4;          // 50000
    const int E = in_sizes[1] / 2;          // 800000
    const int* srcp = edge;
    const int* dstp = edge + E;

    float* h0  = (float*)d_ws;              // N*D
    float* h1  = h0 + (size_t)N * D;        // N*D
    float* agg = h1 + (size_t)N * D;        // N*D
    float* cnt = agg + (size_t)N * D;       // N

    float* fout = (float*)d_out;

    const int WPB = 8;                      // waves per 256-thread block

    // 1) embeddings + layernorm -> h0
    {
        int blocks = (N + WPB - 1) / WPB;
        embed_ln_kernel<<<blocks, 256, 0, stream>>>(x, syn, lem, pos, sen,
                                                    ln_g, ln_b, h0, N);
    }

    // ---- layer 0 ----
    {
        long long zn = (long long)N * D + N;   // agg + cnt contiguous
        long long zb = (zn + 255) / 256;
        zero_kernel<<<(unsigned)zb, 256, 0, stream>>>(agg, zn);

        int blocks = (E + WPB - 1) / WPB;
        scatter_kernel<<<blocks, 256, 0, stream>>>(h0, srcp, dstp, agg, cnt, E, 1);

        long long tot = (long long)N * D;
        normalize_kernel<<<(unsigned)((tot + 255) / 256), 256, 0, stream>>>(agg, cnt, tot);

        sage_gemm_wmma_kernel<<<(N + 15) / 16, 256, 0, stream>>>(agg, h0, Wl0, bl0,
                                                                 Wr0, h1, /*relu=*/1);
    }

    // ---- layer 1 (cnt reused: identical graph) ----
    {
        long long zn = (long long)N * D;
        zero_kernel<<<(unsigned)((zn + 255) / 256), 256, 0, stream>>>(agg, zn);

        int blocks = (E + WPB - 1) / WPB;
        scatter_kernel<<<blocks, 256, 0, stream>>>(h1, srcp, dstp, agg, cnt, E, 0);

        long long tot = (long long)N * D;
        normalize_kernel<<<(unsigned)((tot + 255) / 256), 256, 0, stream>>>(agg, cnt, tot);

        sage_gemm_wmma_kernel<<<(N + 15) / 16, 256, 0, stream>>>(agg, h1, Wl1, bl1,
                                                                 Wr1, fout, /*relu=*/0);
    }
}